// Conv_39041252721264
// MI455X (gfx1250) — compile-verified
//
#include <hip/hip_runtime.h>

// Depthwise 9x9 conv, single shared kernel, fp32, PAD=4 STRIDE=1.
// X: (32,64,256,256) -> 2048 images of 256x256. K: 81 floats. Out: same shape.
//
// Per input row r and 16-col tile, V_WMMA_F32_16X16X4_F32 computes
//   G[u, x] = sum_v K[u,v] * X[r, x0+x+v-4]   (M=u, N=x, K=v in 3 chunks of 4)
// then per-lane VALU accumulates out[y,x] = sum_u G_{r=y+u-4}[u,x].
// A 256-thread block covers a 32-row x 128-col band; the 40x139 halo window
// is staged in LDS (zero-padded) so the WMMA loop is fully branch-free.
// 32-row bands cut the vertical-halo tax to 1.25x reads and 60 WMMA per
// 16x16 tile (vs 1.5x / 72 for 16-row bands).

typedef __attribute__((ext_vector_type(2))) float v2f;
typedef __attribute__((ext_vector_type(8))) float v8f;

namespace {
constexpr int H  = 256;
constexpr int W  = 256;
constexpr int KS = 9;
constexpr int PADR = 4;
constexpr int BAND_ROWS  = 32;                // output rows per block
constexpr int BAND_COLS  = 128;               // 8 waves * 16 columns
constexpr int BANDS_X    = W / BAND_COLS;     // 2
constexpr int BANDS_Y    = H / BAND_ROWS;     // 8
constexpr int LDS_ROWS   = BAND_ROWS + 8;     // 40 = out rows + halo
constexpr int LDS_COLS   = BAND_COLS + 4 + 7; // cx in [-4, +134] rel. band -> 139
constexpr int LDS_STRIDE = 140;               // padded row stride (dwords)
}

__global__ __launch_bounds__(256) void dwconv9x9_wmma(
    const float* __restrict__ X, const float* __restrict__ Kf,
    float* __restrict__ Out)
{
    __shared__ float lds[LDS_ROWS * LDS_STRIDE];

    const int lane = threadIdx.x & 31;
    const int wave = threadIdx.x >> 5;

    // Block -> (image, row band, col band). Grid divides exactly; no early exit.
    const int bc   = blockIdx.x >> 4;               // / (BANDS_Y*BANDS_X) = /16
    const int band = blockIdx.x & 15;
    const int ty   = band >> 1;
    const int bx   = band & 1;
    const int y0      = ty * BAND_ROWS;
    const int blockX0 = bx * BAND_COLS;

    const int n  = lane & 15;   // N index (x within tile); also M index (u) for A
    const int hi = lane >> 4;   // lane half: selects K taps {0,1} vs {2,3}

    const float* img = X + (size_t)bc * (H * W);

    // ---- Stage 40x139 halo window into LDS, branch-free (clamp + select) ----
    for (int idx = threadIdx.x; idx < LDS_ROWS * LDS_STRIDE; idx += 256) {
        const int row = idx / LDS_STRIDE;
        const int col = idx - row * LDS_STRIDE;
        const int gr = y0 - PADR + row;          // global row
        const int gc = blockX0 - 4 + col;        // global col
        const int cr = min(max(gr, 0), H - 1);
        const int cg = min(max(gc, 0), W - 1);
        const bool ok = (gr >= 0) & (gr < H) & (gc >= 0) & (gc < W) &
                        (col < LDS_COLS);
        const float v = img[(size_t)cr * W + cg];
        lds[idx] = ok ? v : 0.0f;
    }

    // ---- A operands (kernel taps), ISA 32-bit A 16x4 layout ----
    // lane = M = u ; V0/.x holds K = vb + (hi?2:0) ; V1/.y = +1. Branch-free.
    v2f a[3];
#pragma unroll
    for (int t = 0; t < 3; ++t) {
        const int v0 = 4 * t + (hi ? 2 : 0);
        const int u  = n;
        const int i0 = min(u * KS + v0,     KS * KS - 1);
        const int i1 = min(u * KS + v0 + 1, KS * KS - 1);
        a[t].x = (u < KS && (v0 + 0) < KS) ? Kf[i0] : 0.0f;
        a[t].y = (u < KS && (v0 + 1) < KS) ? Kf[i1] : 0.0f;
    }

    __syncthreads();

    // Per-lane accumulators: out rows y0+j at column x0 + (lane&15).
    // Lanes 0-15 hold the real sums (u=0..8); upper-lane accs are garbage.
    float acc[BAND_ROWS];
#pragma unroll
    for (int j = 0; j < BAND_ROWS; ++j) acc[j] = 0.0f;

    const int x0 = blockX0 + wave * 16;
    // LDS col 0 == global col blockX0-4, so cx = x+v-4 maps to col x+v.
    const int bcol = wave * 16 + n;   // + v gives the B element column

#pragma unroll
    for (int i = 0; i < LDS_ROWS; ++i) {
        const float* lrow = &lds[i * LDS_STRIDE + bcol];

        v8f c = {};
#pragma unroll
        for (int t = 0; t < 3; ++t) {
            // B operands, ISA 32-bit B 4x16 layout: lane = N = x;
            // lower lanes hold K = {4t,4t+1}, upper K = {4t+2,4t+3}.
            const int v = 4 * t + (hi ? 2 : 0);
            v2f b;
            b.x = lrow[v];
            b.y = lrow[v + 1];
            c = __builtin_amdgcn_wmma_f32_16x16x4_f32(
                    false, a[t], false, b, (short)0, c, false, false);
        }

        // c[g]: lanes0-15 -> G[u=g, x], lanes16-31 -> G[u=g+8, x].
        // Bring the u=8 value (upper-half c[0]) down to the lower half.
        union { float f; int i; } cv;
        cv.f = c[0];
        cv.i = __builtin_amdgcn_ds_swizzle(cv.i, 0x401F);  // SWAPX16
        const float c8 = cv.f;

        // out row y = r + 4 - u  ->  j = i - u  (static indices after unroll)
#pragma unroll
        for (int u = 0; u < KS; ++u) {
            const int j = i - u;
            if (j < 0 || j >= BAND_ROWS) continue;
            acc[j] += (u == 8) ? c8 : c[u];
        }
    }

    // Full-wave stores: lanes 0-15 write row y0+j, lanes 16-31 write row
    // y0+j+16 (data brought up from the lower half with one SWAPX16 each).
    float* outImg = Out + (size_t)bc * (H * W);
    const int orow = y0 + hi * 16;          // per-lane base output row
#pragma unroll
    for (int j = 0; j < 16; ++j) {
        union { float f; int i; } sv;
        sv.f = acc[j + 16];
        sv.i = __builtin_amdgcn_ds_swizzle(sv.i, 0x401F);  // SWAPX16
        const float val = hi ? sv.f : acc[j];
        outImg[(size_t)(orow + j) * W + (x0 + n)] = val;
    }
}

extern "C" void kernel_launch(void* const* d_in, const int* in_sizes, int n_in,
                              void* d_out, int out_size, void* d_ws, size_t ws_size,
                              hipStream_t stream) {
    const float* X  = (const float*)d_in[0];
    const float* Kf = (const float*)d_in[1];
    float* Out      = (float*)d_out;

    const int nImages = in_sizes[0] / (H * W);              // 2048
    const int blocks  = nImages * BANDS_Y * BANDS_X;        // 32768

    dwconv9x9_wmma<<<blocks, 256, 0, stream>>>(X, Kf, Out);
}